// PointNetConv_63797444214962
// MI455X (gfx1250) — compile-verified
//
#include <hip/hip_runtime.h>

typedef float v2f __attribute__((ext_vector_type(2)));
typedef float v8f __attribute__((ext_vector_type(8)));

#define BB 8
#define HH 64
#define WWD 1024
#define OO 32
#define HWW (HH * WWD)

// One wave32 handles 16 consecutive positions (one row chunk) x all 32 channels.
// A-matrix (16x4, f32) = weight tile (channels x features), two tiles for 32 ch.
// B-matrix (4x16, f32) = feats^T (features x positions), rebuilt per neighbor k.
// C-matrix = bias broadcast. D layout: VGPR j -> channel j (+8 for hi half),
// lane&15 -> position, so the per-position mask multiplies all 8 VGPRs directly.
__global__ __launch_bounds__(256) void pconv_wmma_kernel(
    const float* __restrict__ x,  const float* __restrict__ X,
    const float* __restrict__ A,  const float* __restrict__ E,
    const float* __restrict__ M,  const float* __restrict__ cw,
    const float* __restrict__ cb, float* __restrict__ out)
{
    const int tid  = threadIdx.x;
    const int lane = tid & 31;
    const int gw   = blockIdx.x * 8 + (tid >> 5);   // global wave id
    const int base = gw * 16;                       // first position of this wave
    const int b    = base / HWW;
    const int rem  = base - b * HWW;
    const int h    = rem / WWD;
    const int w0   = rem - h * WWD;

    const int p  = lane & 15;   // position within chunk / channel within tile
    const int hi = lane >> 4;   // half-wave select (K0/K1 vs K2/K3)

    // ---- A tiles: weights. lane holds ch=(lane&15)+16*tile, K = 2*hi, 2*hi+1
    const int kk = hi * 2;
    v2f a0, a1;
    a0.x = cw[p * 4 + kk];        a0.y = cw[p * 4 + kk + 1];
    a1.x = cw[(p + 16) * 4 + kk]; a1.y = cw[(p + 16) * 4 + kk + 1];

    // ---- C tiles: bias broadcast (D[m][n] C value depends on channel m only)
    v8f c0, c1;
#pragma unroll
    for (int j = 0; j < 8; ++j) {
        c0[j] = cb[j + 8 * hi];
        c1[j] = cb[16 + j + 8 * hi];
    }

    // ---- center values for this lane's position
    const int cidx = b * HWW + h * WWD + w0 + p;
    const float Xc = X[cidx];
    const float Ac = A[cidx];
    const float Ec = E[cidx];
    const float Mc = M[cidx];

    float acc0[8], acc1[8];
#pragma unroll
    for (int j = 0; j < 8; ++j) { acc0[j] = -3.402823466e38f; acc1[j] = -3.402823466e38f; }

#pragma unroll
    for (int k = 0; k < 9; ++k) {
        const int di = k / 3 - 1;
        const int dj = k % 3 - 1;
        const int hh = h + di;
        const int ww = w0 + p + dj;
        const bool inb = (hh >= 0) && (hh < HH) && (ww >= 0) && (ww < WWD);
        const float inf_f = inb ? 1.0f : 0.0f;          // zero-pad via multiply
        const int hc = min(max(hh, 0), HH - 1);         // clamped addr: no EXEC
        const int wc = min(max(ww, 0), WWD - 1);        // divergence around WMMA
        const int idx = b * HWW + hc * WWD + wc;

        const float ux = x[idx] * inf_f;
        const float uA = A[idx] * inf_f;
        const float uE = E[idx] * inf_f;
        const float uM = M[idx] * inf_f;

        const float Ad = uA - Ac;
        const float Ed = uE - Ec;
        const float cA = __cosf(Ad), sA = __sinf(Ad);
        const float cE = __cosf(Ed), sE = __sinf(Ed);

        const float f0 = ux;
        const float f1 = ux * cA * cE - Xc;
        const float f2 = ux * cA * sE;
        const float f3 = ux * sA;

        // B-matrix 4x16: lanes 0-15 -> {K0,K1}, lanes 16-31 -> {K2,K3}
        v2f bv;
        bv.x = hi ? f2 : f0;
        bv.y = hi ? f3 : f1;

        // D = W_tile(16x4) x feats^T(4x16) + bias
        v8f d0 = __builtin_amdgcn_wmma_f32_16x16x4_f32(
            false, a0, false, bv, (short)0, c0, false, false);
        v8f d1 = __builtin_amdgcn_wmma_f32_16x16x4_f32(
            false, a1, false, bv, (short)0, c1, false, false);

        const float mv = Mc * uM;   // position-dependent only -> per-lane scalar
#pragma unroll
        for (int j = 0; j < 8; ++j) {
            acc0[j] = fmaxf(acc0[j], d0[j] * mv);
            acc1[j] = fmaxf(acc1[j], d1[j] * mv);
        }
    }

    // ---- store: channel = j + 8*hi (+16 for tile1), position = w0 + p
    // consecutive lanes in a half-wave -> consecutive addresses (64B coalesced)
    const int obase = (b * OO) * HWW + h * WWD + w0 + p;
#pragma unroll
    for (int j = 0; j < 8; ++j) {
        out[obase + (j + 8 * hi) * HWW]        = acc0[j];
        out[obase + (16 + j + 8 * hi) * HWW]   = acc1[j];
    }
}

extern "C" void kernel_launch(void* const* d_in, const int* in_sizes, int n_in,
                              void* d_out, int out_size, void* d_ws, size_t ws_size,
                              hipStream_t stream) {
    const float* x  = (const float*)d_in[0];
    const float* X  = (const float*)d_in[1];
    const float* A  = (const float*)d_in[2];
    const float* E  = (const float*)d_in[3];
    const float* M  = (const float*)d_in[4];
    const float* cw = (const float*)d_in[5];
    const float* cb = (const float*)d_in[6];
    float* out = (float*)d_out;

    // 524288 positions / 16 per wave = 32768 waves; 8 waves (256 thr) per block
    const int blocks = (BB * HWW) / (16 * 8);
    pconv_wmma_kernel<<<blocks, 256, 0, stream>>>(x, X, A, E, M, cw, cb, out);
}